// ChannelMoE_2491081032217
// MI455X (gfx1250) — compile-verified
//
#include <hip/hip_runtime.h>
#include <math.h>

// Problem constants (from reference)
#define B_   64
#define C_   128
#define L_   4096
#define K_   4
#define EPS_ 1e-5f

typedef float v2f __attribute__((ext_vector_type(2)));
typedef float v8f __attribute__((ext_vector_type(8)));

// ---------------- wave32 helpers ----------------
__device__ __forceinline__ float wave_sum32(float v) {
    #pragma unroll
    for (int m = 16; m >= 1; m >>= 1) v += __shfl_xor(v, m, 32);
    return v;
}

__device__ __forceinline__ void wave_argmax32(float& v, int& idx) {
    #pragma unroll
    for (int m = 16; m >= 1; m >>= 1) {
        float ov = __shfl_xor(v, m, 32);
        int   oi = __shfl_xor(idx, m, 32);
        if (ov > v || (ov == v && oi < idx)) { v = ov; idx = oi; }
    }
}

// ---------------- Kernel 1: mean over L ----------------
// grid = B*C blocks, 256 threads. Each thread sums 16 elems via 4x float4.
__global__ void moe_pool_kernel(const float* __restrict__ x, float* __restrict__ xpool) {
    const int row = blockIdx.x;                       // b*C + c
    const float4* xr = (const float4*)(x + (size_t)row * L_);
    float s = 0.f;
    #pragma unroll
    for (int it = 0; it < 4; ++it) {
        float4 v = xr[it * 256 + threadIdx.x];
        s += v.x + v.y + v.z + v.w;
    }
    s = wave_sum32(s);
    __shared__ float sw[8];
    if ((threadIdx.x & 31) == 0) sw[threadIdx.x >> 5] = s;
    __syncthreads();
    if (threadIdx.x == 0) {
        float t = 0.f;
        #pragma unroll
        for (int w = 0; w < 8; ++w) t += sw[w];
        xpool[row] = t * (1.0f / (float)L_);
    }
}

// ---------------- Kernel 2: scores (WMMA f32 16x16x4) + LN + top4 + softmax ----
// grid = C blocks (one per channel i), 256 threads = 8 waves.
// Phase 1: scores[b, o] for all 64 batches x 128 outputs via v_wmma_f32_16x16x4_f32.
//   wave w: b-tile = w>>1 (16 batches), o-group = w&1 (4 o-tiles of 16) -> 4 WMMAs.
// Phase 2: wave-per-row LayerNorm + iterative top-4 + softmax; emit (val, idx).
__global__ void moe_scores_kernel(const float* __restrict__ xpool,
                                  const float* __restrict__ w1,
                                  const float* __restrict__ b1,
                                  const float* __restrict__ w2,
                                  const float* __restrict__ b2,
                                  const float* __restrict__ gamma,
                                  const float* __restrict__ beta,
                                  const float* __restrict__ temperature,
                                  float* __restrict__ out_vals,
                                  int* __restrict__ out_idx) {
    __shared__ float s_scores[B_][C_];                 // 64 x 128 f32 = 32 KB

    const int i    = blockIdx.x;                       // channel / group index
    const int wave = threadIdx.x >> 5;                 // 0..7
    const int lane = threadIdx.x & 31;

    // ---- Phase 1: WMMA ----
    {
        const int btile = wave >> 1;                   // 0..3 (batches btile*16..+15)
        const int ogrp  = wave & 1;                    // 0..1 (o tiles 4*ogrp..4*ogrp+3)
        const int m     = lane & 15;                   // matrix row within half-wave
        const int hi    = lane >> 4;                   // 0: K=0,1   1: K=2,3
        const int batch = btile * 16 + m;
        const int k0    = hi * 2;
        const int k1    = k0 + 1;

        // A fragment: A[m, k] = h[batch, i, k] = relu(pool*w1 + b1)
        // ISA 32-bit A 16x4 layout: lanes 0-15 {VGPR0=K0, VGPR1=K1}, lanes 16-31 {K2, K3}
        const float pool = xpool[batch * C_ + i];
        const float a0 = fmaxf(fmaf(pool, w1[i * 4 + k0], b1[i * 4 + k0]), 0.f);
        const float a1 = fmaxf(fmaf(pool, w1[i * 4 + k1], b1[i * 4 + k1]), 0.f);
        v2f A; A.x = a0; A.y = a1;

        #pragma unroll
        for (int q = 0; q < 4; ++q) {
            const int obase = (ogrp * 4 + q) * 16;
            const int o = obase + m;
            // B fragment: B[k, n] = w2[i, o=obase+n, k]  (mirrored half-wave K split)
            v2f Bf;
            Bf.x = w2[(i * C_ + o) * 4 + k0];
            Bf.y = w2[(i * C_ + o) * 4 + k1];
            // C operand carries the bias b2[i, o] (constant down the M dimension)
            const float cb = b2[i * C_ + o];
            v8f Cf = {cb, cb, cb, cb, cb, cb, cb, cb};
            // D = A x B + C  (f32 in/out -> bit-accurate precision class vs reference)
            v8f D = __builtin_amdgcn_wmma_f32_16x16x4_f32(
                false, A, false, Bf, (short)0, Cf, false, false);
            // C/D layout: lanes 0-15: VGPR r -> M=r, N=lane; lanes 16-31: M=8+r, N=lane-16
            #pragma unroll
            for (int r = 0; r < 8; ++r) {
                s_scores[btile * 16 + hi * 8 + r][o] = D[r];
            }
        }
    }
    __syncthreads();

    // ---- Phase 2: per-row LN + top-4 + softmax (wave per row, 8 rows per wave) ----
    const float invT = 1.0f / temperature[0];
    for (int rr = 0; rr < 8; ++rr) {
        const int b = wave * 8 + rr;                   // batch
        // lane holds o = 4*lane .. 4*lane+3
        float4 v4 = ((const float4*)s_scores[b])[lane];
        float v[4] = {v4.x, v4.y, v4.z, v4.w};

        float s = v[0] + v[1] + v[2] + v[3];
        const float mu = wave_sum32(s) * (1.0f / (float)C_);
        float sq = 0.f;
        #pragma unroll
        for (int e = 0; e < 4; ++e) { float d = v[e] - mu; sq += d * d; }
        const float var = wave_sum32(sq) * (1.0f / (float)C_);
        const float inv = rsqrtf(var + EPS_);
        #pragma unroll
        for (int e = 0; e < 4; ++e) {
            const int o = lane * 4 + e;
            v[e] = (fmaf((v[e] - mu) * inv, gamma[o], beta[o])) * invT;
        }

        // iterative top-4 (wave argmax, invalidate winner each round)
        float topv[K_]; int topo[K_];
        #pragma unroll
        for (int k = 0; k < K_; ++k) {
            float lv = -INFINITY; int lo = 0x7FFFFFFF;
            #pragma unroll
            for (int e = 0; e < 4; ++e)
                if (v[e] > lv) { lv = v[e]; lo = lane * 4 + e; }
            wave_argmax32(lv, lo);
            topv[k] = lv; topo[k] = lo;
            if ((lo >> 2) == lane) v[lo & 3] = -INFINITY;   // winner lane clears slot
        }

        // softmax over the 4 winners (topv[0] is the max)
        float ex[K_], den = 0.f;
        #pragma unroll
        for (int k = 0; k < K_; ++k) { ex[k] = __expf(topv[k] - topv[0]); den += ex[k]; }
        const float rden = 1.0f / den;

        if (lane < K_) {
            const size_t base = ((size_t)b * C_ + i) * K_;
            out_vals[base + lane] = ex[lane] * rden;
            out_idx[base + lane]  = topo[lane];
        }
    }
}

// ---------------- Kernel 3: sparse channel mix + residual ----------------
// grid = B*C blocks, 256 threads. out[b,c,l] = x[b,c,l] + sum_k w_k * x[b, idx_k, l]
__global__ void moe_mix_kernel(const float* __restrict__ x,
                               const float* __restrict__ vals,
                               const int* __restrict__ idx,
                               float* __restrict__ out) {
    const int row = blockIdx.x;                       // b*C + c
    const int b = row >> 7;
    const float* xb = x + (size_t)b * C_ * L_;

    const float w0 = vals[row * K_ + 0], w1 = vals[row * K_ + 1];
    const float w2 = vals[row * K_ + 2], w3 = vals[row * K_ + 3];
    const int   i0 = idx[row * K_ + 0],  i1 = idx[row * K_ + 1];
    const int   i2 = idx[row * K_ + 2],  i3 = idx[row * K_ + 3];

    const float4* rid = (const float4*)(x + (size_t)row * L_);      // identity row
    const float4* r0  = (const float4*)(xb + (size_t)i0 * L_);
    const float4* r1  = (const float4*)(xb + (size_t)i1 * L_);
    const float4* r2  = (const float4*)(xb + (size_t)i2 * L_);
    const float4* r3  = (const float4*)(xb + (size_t)i3 * L_);
    float4* o4 = (float4*)(out + (size_t)row * L_);

    #pragma unroll
    for (int it = 0; it < 4; ++it) {
        const int e = it * 256 + threadIdx.x;
        float4 xi = rid[e];
        float4 a = r0[e], bb = r1[e], cc = r2[e], dd = r3[e];
        float4 res;
        res.x = xi.x + w0 * a.x + w1 * bb.x + w2 * cc.x + w3 * dd.x;
        res.y = xi.y + w0 * a.y + w1 * bb.y + w2 * cc.y + w3 * dd.y;
        res.z = xi.z + w0 * a.z + w1 * bb.z + w2 * cc.z + w3 * dd.z;
        res.w = xi.w + w0 * a.w + w1 * bb.w + w2 * cc.w + w3 * dd.w;
        o4[e] = res;
    }
}

extern "C" void kernel_launch(void* const* d_in, const int* in_sizes, int n_in,
                              void* d_out, int out_size, void* d_ws, size_t ws_size,
                              hipStream_t stream) {
    const float* x     = (const float*)d_in[0];   // [B,C,L]
    const float* w1    = (const float*)d_in[1];   // [C,4]
    const float* b1    = (const float*)d_in[2];   // [C,4]
    const float* w2    = (const float*)d_in[3];   // [C,C,4]
    const float* b2    = (const float*)d_in[4];   // [C,C]
    const float* gamma = (const float*)d_in[5];   // [C]
    const float* beta  = (const float*)d_in[6];   // [C]
    const float* temp  = (const float*)d_in[7];   // [1]
    float* out = (float*)d_out;

    // workspace layout
    char* ws = (char*)d_ws;
    float* xpool = (float*)ws;                                   // B*C floats   (32 KB)
    float* vals  = (float*)(ws + ((size_t)B_ * C_ * 4 + 256) * sizeof(float)); // B*C*K
    int*   idxs  = (int*)(vals + (size_t)B_ * C_ * K_);          // B*C*K ints

    moe_pool_kernel<<<B_ * C_, 256, 0, stream>>>(x, xpool);
    moe_scores_kernel<<<C_, 256, 0, stream>>>(xpool, w1, b1, w2, b2,
                                              gamma, beta, temp, vals, idxs);
    moe_mix_kernel<<<B_ * C_, 256, 0, stream>>>(x, vals, idxs, out);
}